// GCNTeacher_39857296506997
// MI455X (gfx1250) — compile-verified
//
#include <hip/hip_runtime.h>

// LightGCN-style propagation for MI455X (gfx1250, wave32).
// out layout (floats): [hsum | h0 | h1 | h2 | h3], each N_NODES*DIM.
// Whole working set (~134 MB: 76.8 MB features + 57.6 MB edges) is L2-resident
// (192 MB L2), so the bound is L2 gather/atomic throughput, not HBM.

#define N_USERS    100000
#define N_ITEMS_P1 50001
#define N_NODES    150001
#define DIM        64
#define ND         ((long)N_NODES * DIM)   /* 9,600,064 */
#define ND4        (ND / 4)                /* 2,400,016 float4 */

// ---------------------------------------------------------------------------
// spmm: h_out[r] += vals[e] * h_prev[c]  (atomic scatter-add)
// One wave32 per 32-edge chunk; lane owns dims [2*lane, 2*lane+1].
// Edge metadata staged global->LDS with async loads, double-buffered per wave
// (each wave consumes only the slots its own lanes staged -> no barriers,
// only s_wait_asynccnt). Accumulation via hardware global_atomic_add_f32 at
// device scope (L2-side f32 adds, no CAS loop).
// ---------------------------------------------------------------------------
__global__ __launch_bounds__(256) void gcn_spmm_kernel(
    const float* __restrict__ h_prev, float* __restrict__ h_out,
    const int* __restrict__ rows, const int* __restrict__ cols,
    const float* __restrict__ vals, int nedges) {
  __shared__ int   s_row[2][256];
  __shared__ int   s_col[2][256];
  __shared__ float s_val[2][256];

  const int tid   = threadIdx.x;
  const int lane  = tid & 31;
  const int wslot = tid & ~31;   // this wave's base slot in the LDS tiles

  const long nchunks = ((long)nedges + 31) / 32;
  const long wave_id = (long)blockIdx.x * (blockDim.x >> 5) + (tid >> 5);
  const long nwaves  = (long)gridDim.x * (blockDim.x >> 5);

  const unsigned long long rows_base = (unsigned long long)rows;
  const unsigned long long cols_base = (unsigned long long)cols;
  const unsigned long long vals_base = (unsigned long long)vals;
  const unsigned long long hout_base = (unsigned long long)h_out;

  auto stage = [&](int p, long chunk) {
    long e = chunk * 32 + lane;
    if (e >= nedges) e = nedges - 1;          // clamp: safe duplicate read
    const unsigned voff  = (unsigned)(e * 4); // byte offset, < 2^31
    // Flat address of an LDS slot truncates to the LDS byte address (ISA 10.2).
    const unsigned a_row = (unsigned)(unsigned long long)&s_row[p][tid];
    const unsigned a_col = (unsigned)(unsigned long long)&s_col[p][tid];
    const unsigned a_val = (unsigned)(unsigned long long)&s_val[p][tid];
    asm volatile("global_load_async_to_lds_b32 %0, %1, %2"
                 :: "v"(a_row), "v"(voff), "s"(rows_base) : "memory");
    asm volatile("global_load_async_to_lds_b32 %0, %1, %2"
                 :: "v"(a_col), "v"(voff), "s"(cols_base) : "memory");
    asm volatile("global_load_async_to_lds_b32 %0, %1, %2"
                 :: "v"(a_val), "v"(voff), "s"(vals_base) : "memory");
  };

  long chunk = wave_id;
  if (chunk >= nchunks) return;
  int p = 0;
  stage(p, chunk);

  for (; chunk < nchunks; chunk += nwaves) {
    const long nxt = chunk + nwaves;
    if (nxt < nchunks) {
      stage(p ^ 1, nxt);                                   // prefetch next tile
      asm volatile("s_wait_asynccnt 3" ::: "memory");      // current tile done
    } else {
      asm volatile("s_wait_asynccnt 0" ::: "memory");
    }

    long rem = (long)nedges - chunk * 32;
    const int cnt = (rem < 32) ? (int)rem : 32;

    // Batch 8 edges: 8 independent b64 gathers in flight, then 16 atomics.
    for (int i0 = 0; i0 < cnt; i0 += 8) {
      const int m = ((cnt - i0) < 8) ? (cnt - i0) : 8;
      unsigned voffs[8];
      float    xs[8], ys[8];
      #pragma unroll
      for (int j = 0; j < 8; ++j) {
        if (j < m) {
          const int   r = s_row[p][wslot + i0 + j];   // LDS broadcast reads
          const int   c = s_col[p][wslot + i0 + j];
          const float v = s_val[p][wslot + i0 + j];
          const float2 hv = *reinterpret_cast<const float2*>(
              h_prev + (size_t)c * DIM + lane * 2);
          xs[j] = v * hv.x;
          ys[j] = v * hv.y;
          voffs[j] = ((unsigned)r * DIM + (unsigned)(lane * 2)) * 4u;
        }
      }
      #pragma unroll
      for (int j = 0; j < 8; ++j) {
        if (j < m) {
          // Fire-and-forget f32 adds at device scope (L2 atomic units).
          // No memory clobber: h_out is never read in this kernel, so the
          // compiler may keep pipelining the next batch's gathers.
          asm volatile("global_atomic_add_f32 %0, %1, %2 scope:SCOPE_DEV"
                       :: "v"(voffs[j]), "v"(xs[j]), "s"(hout_base));
          asm volatile("global_atomic_add_f32 %0, %1, %2 offset:4 scope:SCOPE_DEV"
                       :: "v"(voffs[j]), "v"(ys[j]), "s"(hout_base));
        }
      }
    }
    p ^= 1;
  }
}

// ---------------------------------------------------------------------------
// init: h0 = concat(user_emb, item_emb); zero h1..h3
// ---------------------------------------------------------------------------
__global__ __launch_bounds__(256) void gcn_init_kernel(
    const float4* __restrict__ user4, const float4* __restrict__ item4,
    float* __restrict__ out) {
  float4* h0 = reinterpret_cast<float4*>(out + 1L * ND);
  float4* hz = reinterpret_cast<float4*>(out + 2L * ND);  // h1..h3, 3*ND4
  const long ucnt  = (long)N_USERS * DIM / 4;
  const long total = 4L * ND4;
  const long stride = (long)gridDim.x * blockDim.x;
  for (long i = (long)blockIdx.x * blockDim.x + threadIdx.x; i < total; i += stride) {
    if (i < ND4) {
      h0[i] = (i < ucnt) ? user4[i] : item4[i - ucnt];
    } else {
      hz[i - ND4] = make_float4(0.f, 0.f, 0.f, 0.f);
    }
  }
}

// ---------------------------------------------------------------------------
// sum: hsum = h0 + h1 + h2 + h3
// ---------------------------------------------------------------------------
__global__ __launch_bounds__(256) void gcn_sum_kernel(float* __restrict__ out) {
  const float4* h0 = reinterpret_cast<const float4*>(out + 1L * ND);
  const float4* h1 = reinterpret_cast<const float4*>(out + 2L * ND);
  const float4* h2 = reinterpret_cast<const float4*>(out + 3L * ND);
  const float4* h3 = reinterpret_cast<const float4*>(out + 4L * ND);
  float4* hs = reinterpret_cast<float4*>(out);
  const long stride = (long)gridDim.x * blockDim.x;
  for (long i = (long)blockIdx.x * blockDim.x + threadIdx.x; i < ND4; i += stride) {
    const float4 a = h0[i], b = h1[i], c = h2[i], d = h3[i];
    hs[i] = make_float4(a.x + b.x + c.x + d.x, a.y + b.y + c.y + d.y,
                        a.z + b.z + c.z + d.z, a.w + b.w + c.w + d.w);
  }
}

// ---------------------------------------------------------------------------
extern "C" void kernel_launch(void* const* d_in, const int* in_sizes, int n_in,
                              void* d_out, int out_size, void* d_ws, size_t ws_size,
                              hipStream_t stream) {
  (void)n_in; (void)out_size; (void)d_ws; (void)ws_size;
  const float* user_emb = (const float*)d_in[0];
  const float* item_emb = (const float*)d_in[1];
  const float* vals     = (const float*)d_in[2];
  const int*   rows     = (const int*)d_in[3];
  const int*   cols     = (const int*)d_in[4];
  const int    nedges   = in_sizes[2];
  float* out = (float*)d_out;

  gcn_init_kernel<<<4096, 256, 0, stream>>>(
      (const float4*)user_emb, (const float4*)item_emb, out);

  for (int l = 0; l < 3; ++l) {
    const float* hp = out + (size_t)(1 + l) * ND;
    float*       ho = out + (size_t)(2 + l) * ND;
    gcn_spmm_kernel<<<2048, 256, 0, stream>>>(hp, ho, rows, cols, vals, nedges);
  }

  gcn_sum_kernel<<<2048, 256, 0, stream>>>(out);
}